// SpatialMaskingModule_59493886984281
// MI455X (gfx1250) — compile-verified
//
#include <hip/hip_runtime.h>
#include <hip/hip_bf16.h>
#include <math.h>

// ---------------------------------------------------------------------------
// SpatialMasking: per batch row, distance of each residue Ca to ligand
// centroid, exact top-k nearest selection via LDS radix-select.
//
// Matrix-pipe distance (full f32, V_WMMA_F32_16X16X4_F32):
//   A (16x4)  row_i = [x_i, y_i, z_i, ||p_i||^2]   (pp via one shfl_xor)
//   B (4x16)  every column = [-2cx, -2cy, -2cz, 1]
//   C = 0
//   D row_i (all columns) = ||p_i||^2 - 2 p_i.c
// Raw D stored to LDS (2x ds_store_b128 per wave-tile); a lane-parallel pass
// finishes: key = sqrt(raw + ||c||^2 + eps) + (1-mask)*INF.
// ---------------------------------------------------------------------------

typedef __attribute__((ext_vector_type(2))) float v2f;
typedef __attribute__((ext_vector_type(8))) float v8f;

#define HAVE_WMMA_F32X4 __has_builtin(__builtin_amdgcn_wmma_f32_16x16x4_f32)

#define N_RES   16384
#define N_ATOM  64
#define INF_VAL 1.0e10f
#define EPS_VAL 1e-12f
#define NTHREADS 512

// Dynamic LDS layout (bytes):
//   keys     [0,      65536)   16384 x u32 (raw d2-partial, then key bits)
//   hist     [65536,  66560)   256 x u32 radix histogram
//   astage   [66560,  67584)   64 x 4 floats (atom x,y,z,mask staging)
//   cdata    [67584,  67600)   cx, cy, cz, c.c
//   scal     [67600,  67632)   [0]=prefix/T [1]=krem/R [2]=cless [3]=istar
//   chunk_eq [67632,  67760)   32 x u32 per-chunk equal-key counts
#define SMEM_BYTES (65536 + 1024 + 1024 + 16 + 32 + 128)

__global__ __launch_bounds__(NTHREADS)
void spatial_mask_kernel(const float* __restrict__ pos,     // [B,N,3]
                         const float* __restrict__ rmask,   // [B,N]
                         const float* __restrict__ apos,    // [B,A,3]
                         const float* __restrict__ amask,   // [B,A]
                         const int*   __restrict__ topk_p,  // [1]
                         float* __restrict__ out0,          // [B,N]
                         float* __restrict__ out1)          // [B,N]
{
  extern __shared__ unsigned char smem[];
  unsigned int* keys     = (unsigned int*)(smem);
  float*        keysF    = (float*)       (smem);
  unsigned int* hist     = (unsigned int*)(smem + 65536);
  float*        astage   = (float*)       (smem + 66560);
  float*        cdata    = (float*)       (smem + 67584);
  unsigned int* scal     = (unsigned int*)(smem + 67600);
  unsigned int* chunk_eq = (unsigned int*)(smem + 67632);

  const int b   = blockIdx.x;
  const int tid = threadIdx.x;

  const float* posb   = pos   + (size_t)b * N_RES * 3;
  const float* maskb  = rmask + (size_t)b * N_RES;
  const float* aposb  = apos  + (size_t)b * N_ATOM * 3;
  const float* amaskb = amask + (size_t)b * N_ATOM;

  // ---- centroid (deterministic: parallel stage, serial reduce) ----
  if (tid < N_ATOM) {
    astage[tid * 4 + 0] = aposb[tid * 3 + 0];
    astage[tid * 4 + 1] = aposb[tid * 3 + 1];
    astage[tid * 4 + 2] = aposb[tid * 3 + 2];
    astage[tid * 4 + 3] = amaskb[tid];
  }
  __syncthreads();
  if (tid == 0) {
    float sx = 0.f, sy = 0.f, sz = 0.f, sm = 0.f;
    for (int a = 0; a < N_ATOM; ++a) {
      sx += astage[a * 4 + 0];
      sy += astage[a * 4 + 1];
      sz += astage[a * 4 + 2];
      sm += astage[a * 4 + 3];
    }
    const float inv = 1.0f / sm;
    const float cx = sx * inv, cy = sy * inv, cz = sz * inv;
    cdata[0] = cx; cdata[1] = cy; cdata[2] = cz;
    cdata[3] = cx * cx + cy * cy + cz * cz;
  }
  __syncthreads();
  const float cx = cdata[0], cy = cdata[1], cz = cdata[2], cc = cdata[3];

  // ---- raw d2-partials into LDS via WMMA ----
#if HAVE_WMMA_F32X4
  {
    const int lane   = tid & 31;
    const int wave   = tid >> 5;
    const int nwaves = NTHREADS >> 5;
    const int hi     = lane >> 4;   // 0: K=0,1 half  1: K=2,3 half
    const int r      = lane & 15;   // row within 16x4 A tile

    v2f bv;                         // B 4x16: VGPR0 = K{0,2}, VGPR1 = K{1,3}
    bv.x = hi ? (-2.0f * cz) : (-2.0f * cx);
    bv.y = hi ? 1.0f         : (-2.0f * cy);
    const v8f czero = {};

    for (int tile = wave; tile < N_RES / 16; tile += nwaves) {
      const int rbase = tile * 16;
      const float* prow = posb + (size_t)(rbase + r) * 3;

      const float px = prow[hi * 2];          // x (hi=0) or z (hi=1)
      const float py = prow[1];               // y
      // partial ||p||^2: lane r -> x^2+y^2 ; lane r+16 -> z^2
      const float t    = hi ? (px * px) : (px * px + py * py);
      const float txor = __shfl_xor(t, 16, 32);

      v2f a;                                  // A row: [x,y] or [z, ||p||^2]
      a.x = px;
      a.y = hi ? (t + txor) : py;

      v8f d = __builtin_amdgcn_wmma_f32_16x16x4_f32(
          false, a, false, bv, (short)0, czero, false, false);

      // All columns identical: lane 0 holds rows 0..7, lane 16 rows 8..15.
      // D accumulator = 8 consecutive VGPRs -> vector LDS store.
      if (r == 0) {
        *(v8f*)(keysF + rbase + hi * 8) = d;
      }
    }
  }
  __syncthreads();
  // ---- lane-parallel finalize: raw -> sortable key bits ----
  {
    const float addc = cc + EPS_VAL;          // d2 + eps = raw + cc + eps
    for (int i = tid; i < N_RES; i += NTHREADS) {
      const float raw  = keysF[i];
      const float m    = maskb[i];
      const float dist = __builtin_amdgcn_sqrtf(raw + addc)
                       + (1.0f - m) * INF_VAL;
      keys[i] = __float_as_uint(dist);
    }
  }
#else
  for (int i = tid; i < N_RES; i += NTHREADS) {
    const float* p = posb + (size_t)i * 3;
    const float dx = cx - p[0], dy = cy - p[1], dz = cz - p[2];
    const float d2 = dx * dx + dy * dy + dz * dz;
    const float m  = maskb[i];
    const float dist = __builtin_amdgcn_sqrtf(d2 + EPS_VAL)
                     + (1.0f - m) * INF_VAL;
    keys[i] = __float_as_uint(dist);
  }
#endif
  __syncthreads();

  // ---- exact radix-select of the K-th smallest key (keys are positive
  //      floats, so u32 bit order == float order) ----
  int K = *topk_p;
  if (K > N_RES) K = N_RES;
  if (tid == 0) { scal[0] = 0u; scal[1] = (unsigned)K; scal[2] = 0u; scal[3] = 0u; }
  __syncthreads();

  if (K > 0) {
    for (int pass = 3; pass >= 0; --pass) {
      if (tid < 256) hist[tid] = 0u;
      __syncthreads();
      const unsigned long long prefix = scal[0];
      const int sh = 8 * pass;
      const unsigned long long phi = prefix >> (sh + 8);  // 64-bit: safe at sh=24
      for (int i = tid; i < N_RES; i += NTHREADS) {
        const unsigned long long k = keys[i];
        if ((k >> (sh + 8)) == phi)
          atomicAdd(&hist[(unsigned)(k >> sh) & 255u], 1u);
      }
      __syncthreads();
      if (tid == 0) {
        const unsigned krem = scal[1];
        unsigned cum = 0, bsel = 255;
        for (unsigned bkt = 0; bkt < 256; ++bkt) {
          const unsigned c = hist[bkt];
          if (cum + c >= krem) { bsel = bkt; break; }
          cum += c;
        }
        scal[0] = (unsigned)prefix | (bsel << sh);
        scal[1] = krem - cum;     // rank within the equal-key group
        scal[2] += cum;           // strictly-smaller count
      }
      __syncthreads();
    }

    // ---- deterministic tie-break: index of the R-th key equal to T ----
    if (tid < 32) chunk_eq[tid] = 0u;
    __syncthreads();
    const unsigned T = scal[0];
    for (int i = tid; i < N_RES; i += NTHREADS)
      if (keys[i] == T) atomicAdd(&chunk_eq[i / NTHREADS], 1u);
    __syncthreads();
    if (tid == 0) {
      const unsigned R = scal[1];
      unsigned cum = 0;
      int c0 = 0;
      for (; c0 < 32; ++c0) {
        if (cum + chunk_eq[c0] >= R) break;
        cum += chunk_eq[c0];
      }
      unsigned need  = R - cum;
      unsigned istar = 0;
      for (int i = c0 * NTHREADS; i < (c0 + 1) * NTHREADS; ++i) {
        if (keys[i] == T) {
          if (--need == 0) { istar = (unsigned)i; break; }
        }
      }
      scal[3] = istar;
    }
    __syncthreads();
  }

  // ---- write masks (coalesced) ----
  const unsigned T     = scal[0];
  const unsigned Istar = scal[3];
  const bool dosel     = (K > 0);
  float* o0 = out0 + (size_t)b * N_RES;
  float* o1 = out1 + (size_t)b * N_RES;
  for (int i = tid; i < N_RES; i += NTHREADS) {
    const unsigned k = keys[i];
    const bool sel = dosel && ((k < T) || (k == T && (unsigned)i <= Istar));
    const float m = maskb[i];
    o0[i] = sel ? 0.0f  : m;
    o1[i] = sel ? 32.0f : (1.0f - m);
  }
}

extern "C" void kernel_launch(void* const* d_in, const int* in_sizes, int n_in,
                              void* d_out, int out_size, void* d_ws, size_t ws_size,
                              hipStream_t stream) {
  const float* pos    = (const float*)d_in[0];  // [B,N,3]
  const float* rmask  = (const float*)d_in[1];  // [B,N]
  const float* apos   = (const float*)d_in[2];  // [B,A,3]
  const float* amask  = (const float*)d_in[3];  // [B,A]
  // d_in[4] = max_p (unused by the reference computation)
  const int*   topk   = (const int*)d_in[5];

  const int BN = in_sizes[1];       // B * N
  const int B  = BN / N_RES;

  float* out0 = (float*)d_out;
  float* out1 = out0 + (size_t)BN;

  spatial_mask_kernel<<<B, NTHREADS, SMEM_BYTES, stream>>>(
      pos, rmask, apos, amask, topk, out0, out1);
}